// SequentialModel_76742475644997
// MI455X (gfx1250) — compile-verified
//
#include <hip/hip_runtime.h>
#include <math.h>

typedef float v2f __attribute__((ext_vector_type(2)));
typedef float v8f __attribute__((ext_vector_type(8)));

#define BB   256
#define TT   1024
#define IN   252
#define H    10
#define NL   5
#define G4   40    // 4*H gates
#define NPAD 48    // padded gate width (3 WMMA n-tiles)
#define KC   20    // concat K (input h + recurrent h)

__device__ __forceinline__ float sig_(float x) {
    return 1.0f / (1.0f + __expf(-x));
}
__device__ __forceinline__ float tanh_(float x) {
    return 2.0f * sig_(2.0f * x) - 1.0f;   // tanh via fast sigmoid
}

// ---------------------------------------------------------------------------
// Kernel 1: G0[r, 0..39] = X[r, :] @ Wih0^T + (b_ih0 + b_hh0),  r = b*T + t
// fp32 WMMA 16x16x4, K = 252 = 63 steps of 4. One wave per 16-row M tile.
// Weights live in LDS transposed as Wlds[n][k] so B fragments are b64 loads;
// row 40 is a zero row for the padded n in [40,48).
// ---------------------------------------------------------------------------
__launch_bounds__(512)
__global__ void lstm_l0_gemm(const float* __restrict__ X,
                             const float* __restrict__ Wih0,   // [40][252]
                             const float* __restrict__ bih0,   // [40]
                             const float* __restrict__ bhh0,   // [40]
                             float* __restrict__ G0)           // [B*T][40]
{
    __shared__ float Wlds[41 * IN];

    const int tid = threadIdx.x;
    for (int i = tid; i < 41 * IN; i += blockDim.x)
        Wlds[i] = (i < G4 * IN) ? Wih0[i] : 0.0f;
    __syncthreads();

    const int wave = blockIdx.x * (blockDim.x >> 5) + (tid >> 5);
    const int lane = tid & 31;
    const int m    = lane & 15;
    const int hi   = lane >> 4;
    const int row0 = wave * 16;   // grid sized exactly: (B*T)/16 waves

    // accumulator init = bias (broadcast over the 8 M-rows of each lane)
    v8f acc0, acc1, acc2;
    {
        const float b0 = bih0[m]      + bhh0[m];
        const float b1 = bih0[16 + m] + bhh0[16 + m];
        const float b2 = (32 + m < G4) ? (bih0[32 + m] + bhh0[32 + m]) : 0.0f;
#pragma unroll
        for (int v = 0; v < 8; ++v) { acc0[v] = b0; acc1[v] = b1; acc2[v] = b2; }
    }

    const float2* arow = reinterpret_cast<const float2*>(X + (size_t)(row0 + m) * IN);
    const int n2 = (32 + m < G4) ? (32 + m) : G4;                 // zero row for pad
    const float2* b0p = reinterpret_cast<const float2*>(&Wlds[(size_t)m        * IN]);
    const float2* b1p = reinterpret_cast<const float2*>(&Wlds[(size_t)(16 + m) * IN]);
    const float2* b2p = reinterpret_cast<const float2*>(&Wlds[(size_t)n2       * IN]);

    for (int s = 0; s < IN / 4; ++s) {
        const int e = s * 2 + hi;          // float2 index => element k = s*4 + hi*2
        float2 af = arow[e];
        float2 w0 = b0p[e], w1 = b1p[e], w2 = b2p[e];
        v2f a;  a[0] = af.x; a[1] = af.y;
        v2f b0; b0[0] = w0.x; b0[1] = w0.y;
        v2f b1; b1[0] = w1.x; b1[1] = w1.y;
        v2f b2; b2[0] = w2.x; b2[1] = w2.y;
        acc0 = __builtin_amdgcn_wmma_f32_16x16x4_f32(false, a, false, b0, (short)0, acc0, false, false);
        acc1 = __builtin_amdgcn_wmma_f32_16x16x4_f32(false, a, false, b1, (short)0, acc1, false, false);
        acc2 = __builtin_amdgcn_wmma_f32_16x16x4_f32(false, a, false, b2, (short)0, acc2, false, false);
    }

    float* gout = G0 + (size_t)row0 * G4;
#pragma unroll
    for (int v = 0; v < 8; ++v) {
        const int r = v + 8 * hi;
        gout[(size_t)r * G4 + m]      = acc0[v];
        gout[(size_t)r * G4 + 16 + m] = acc1[v];
        if (32 + m < G4) gout[(size_t)r * G4 + 32 + m] = acc2[v];
    }
}

// ---------------------------------------------------------------------------
// Kernel 2: recurrent part. One workgroup (1 wave) per 16-row batch tile.
// Per layer: gates[16][40] = Cinit + A[16][20] @ B[20][40] via fp32 WMMA,
//   A = concat(h_{l-1}^new, h_l^old)  (layer 0: concat(h_0^old, zeros),
//       with C init pulled from the precomputed G0 which already has bias).
// Activations via LDS round-trip; FC+sigmoid fused per timestep.
// ---------------------------------------------------------------------------
__launch_bounds__(32)
__global__ void lstm_recurrent(const float* __restrict__ G0,     // [B*T][40]
                               const float* __restrict__ whh0,   // [40][10]
                               const float* __restrict__ wihr,   // [4][40][10]
                               const float* __restrict__ whhr,   // [4][40][10]
                               const float* __restrict__ bihr,   // [4][40]
                               const float* __restrict__ bhhr,   // [4][40]
                               const float* __restrict__ fcw,    // [10]
                               const float* __restrict__ fcb,    // [1]
                               float* __restrict__ out)          // [B*T]
{
    __shared__ float Wl[NL][NPAD][KC];   // transposed: Wl[l][n][k], b64-friendly
    __shared__ float bias[NL][NPAD];
    __shared__ float hst[NL][16][H];
    __shared__ float cst[NL][16][H];
    __shared__ float gates[16][NPAD];

    const int lane = threadIdx.x;
    const int b0   = blockIdx.x * 16;

    // --- init packed weights / biases / state -----------------------------
    for (int idx = lane; idx < NL * NPAD * KC; idx += 32) {
        const int l   = idx / (NPAD * KC);
        const int rem = idx % (NPAD * KC);
        const int n   = rem / KC;
        const int k   = rem % KC;
        float w = 0.0f;
        if (n < G4) {
            if (l == 0) {
                if (k < H) w = whh0[n * H + k];                    // rows 10..19 = 0
            } else {
                if (k < H) w = wihr[((l - 1) * G4 + n) * H + k];
                else       w = whhr[((l - 1) * G4 + n) * H + (k - H)];
            }
        }
        (&Wl[0][0][0])[idx] = w;
    }
    for (int idx = lane; idx < NL * NPAD; idx += 32) {
        const int l = idx / NPAD, n = idx % NPAD;
        float b = 0.0f;
        if (l >= 1 && n < G4) b = bihr[(l - 1) * G4 + n] + bhhr[(l - 1) * G4 + n];
        (&bias[0][0])[idx] = b;
    }
    for (int idx = lane; idx < NL * 16 * H; idx += 32) {
        (&hst[0][0][0])[idx] = 0.0f;
        (&cst[0][0][0])[idx] = 0.0f;
    }
    float fw[H];
#pragma unroll
    for (int j = 0; j < H; ++j) fw[j] = fcw[j];
    const float fb = fcb[0];
    __syncthreads();

    const int m  = lane & 15;
    const int hi = lane >> 4;

#pragma unroll 1
    for (int t = 0; t < TT; ++t) {
#pragma unroll 1
        for (int l = 0; l < NL; ++l) {
            // ---- C init: layer 0 from G0 (bias already folded), else bias
            v8f acc0, acc1, acc2;
            if (l == 0) {
#pragma unroll
                for (int v = 0; v < 8; ++v) {
                    const float* gr = G0 + (((size_t)(b0 + v + 8 * hi)) * TT + t) * G4;
                    acc0[v] = gr[m];
                    acc1[v] = gr[16 + m];
                    acc2[v] = (32 + m < G4) ? gr[32 + m] : 0.0f;
                }
            } else {
                const float z0 = bias[l][m], z1 = bias[l][16 + m], z2 = bias[l][32 + m];
#pragma unroll
                for (int v = 0; v < 8; ++v) { acc0[v] = z0; acc1[v] = z1; acc2[v] = z2; }
            }

            const float* inpH = (l == 0) ? &hst[0][0][0] : &hst[l - 1][0][0];
            const float* curH = &hst[l][0][0];

            // ---- K loop: 5 steps of WMMA f32 16x16x4 over K = 20
#pragma unroll
            for (int s = 0; s < 5; ++s) {
                const int k0 = s * 4 + hi * 2;
                float a0, a1;
                {
                    int k = k0;
                    a0 = (k < H) ? inpH[m * H + k]
                                 : ((l == 0) ? 0.0f : curH[m * H + (k - H)]);
                    k = k0 + 1;
                    a1 = (k < H) ? inpH[m * H + k]
                                 : ((l == 0) ? 0.0f : curH[m * H + (k - H)]);
                }
                v2f a;  a[0] = a0; a[1] = a1;
                v2f b0; b0[0] = Wl[l][m][k0];      b0[1] = Wl[l][m][k0 + 1];
                v2f b1; b1[0] = Wl[l][16 + m][k0]; b1[1] = Wl[l][16 + m][k0 + 1];
                v2f b2; b2[0] = Wl[l][32 + m][k0]; b2[1] = Wl[l][32 + m][k0 + 1];
                acc0 = __builtin_amdgcn_wmma_f32_16x16x4_f32(false, a, false, b0, (short)0, acc0, false, false);
                acc1 = __builtin_amdgcn_wmma_f32_16x16x4_f32(false, a, false, b1, (short)0, acc1, false, false);
                acc2 = __builtin_amdgcn_wmma_f32_16x16x4_f32(false, a, false, b2, (short)0, acc2, false, false);
            }

            // ---- spill D fragments to LDS (layout: M = v + 8*hi, N = lane&15)
#pragma unroll
            for (int v = 0; v < 8; ++v) {
                const int r = v + 8 * hi;
                gates[r][m]      = acc0[v];
                gates[r][16 + m] = acc1[v];
                gates[r][32 + m] = acc2[v];
            }

            // ---- elementwise LSTM update: 16 rows x 10 units, 5 per lane
#pragma unroll
            for (int q = 0; q < 5; ++q) {
                const int idx = lane + q * 32;      // 0..159
                const int r = idx / H;
                const int j = idx % H;
                const float ig = gates[r][j];
                const float fg = gates[r][H + j];
                const float gg = gates[r][2 * H + j];
                const float og = gates[r][3 * H + j];
                const float cn = sig_(fg) * cst[l][r][j] + sig_(ig) * tanh_(gg);
                const float hn = sig_(og) * tanh_(cn);
                cst[l][r][j] = cn;
                hst[l][r][j] = hn;
            }
        }

        // ---- fused FC(10->1) + sigmoid on top-layer h
        if (lane < 16) {
            float d = fb;
#pragma unroll
            for (int j = 0; j < H; ++j) d += hst[NL - 1][lane][j] * fw[j];
            out[(size_t)(b0 + lane) * TT + t] = sig_(d);
        }
    }
}

extern "C" void kernel_launch(void* const* d_in, const int* in_sizes, int n_in,
                              void* d_out, int out_size, void* d_ws, size_t ws_size,
                              hipStream_t stream) {
    const float* x    = (const float*)d_in[0];
    const float* wih0 = (const float*)d_in[1];
    const float* whh0 = (const float*)d_in[2];
    const float* bih0 = (const float*)d_in[3];
    const float* bhh0 = (const float*)d_in[4];
    const float* wihr = (const float*)d_in[5];
    const float* whhr = (const float*)d_in[6];
    const float* bihr = (const float*)d_in[7];
    const float* bhhr = (const float*)d_in[8];
    const float* fcw  = (const float*)d_in[9];
    const float* fcb  = (const float*)d_in[10];
    float* out = (float*)d_out;
    float* G0  = (float*)d_ws;          // B*T*40 floats = 41.9 MB

    // Kernel 1: 16384 M-tiles, 16 waves (512 threads) per block -> 1024 blocks
    const int mtiles = (BB * TT) / 16;
    hipLaunchKernelGGL(lstm_l0_gemm, dim3(mtiles / 16), dim3(512), 0, stream,
                       x, wih0, bih0, bhh0, G0);

    // Kernel 2: one wave per 16-row batch tile -> 16 workgroups
    hipLaunchKernelGGL(lstm_recurrent, dim3(BB / 16), dim3(32), 0, stream,
                       G0, whh0, wihr, whhr, bihr, bhhr, fcw, fcb, out);
}